// GroupNorm_89739046683555
// MI455X (gfx1250) — compile-verified
//
#include <hip/hip_runtime.h>
#include <hip/hip_bf16.h>

// ---------------------------------------------------------------------------
// Segment GroupNorm, MI455X (gfx1250).
// Memory-bound: 512MB x read twice + 512MB out => ~66us floor @ 23.3 TB/s.
// Pass1: async-copy 16x64 f32 tiles to LDS (ASYNCcnt), reduce rows with
//        V_WMMA_F32_16X16X4_F32 against a ones matrix (sum & sumsq chains),
//        accumulate per-(segment,group) in LDS, flush with global atomics.
// Pass2: tiny finalize (mean, rstd).
// Pass3: float4 streaming normalize (bandwidth optimal).
// ---------------------------------------------------------------------------

#define N_ROWS 500000
#define CCH    256
#define G      4
#define GC     64
#define BSEG   64
#define BG     (BSEG * G)
#define EPS_F  1e-5f

#define TILE_R     16
#define LDS_PITCH  68          // floats/row: 272B = 16B-aligned, +4 bank shift/row
#define WPB        8           // waves per block (256 threads, wave32)

typedef __attribute__((ext_vector_type(2))) float v2f;
typedef __attribute__((ext_vector_type(8))) float v8f;

#if defined(__has_builtin)
#if __has_builtin(__builtin_amdgcn_wmma_f32_16x16x4_f32)
#define HAVE_WMMA_F32X4 1
#endif
#endif

// ws layout (floats): [0,BG) sum | [BG,2BG) sumsq | [2BG,2BG+B) deg
//                     | then BG mean | then BG rstd
__global__ void gn_zero(float* __restrict__ ws) {
    const int t = threadIdx.x;
    if (t < 2 * BG + BSEG) ws[t] = 0.0f;
}

__global__ __launch_bounds__(256) void gn_stats(const float* __restrict__ x,
                                                const int* __restrict__ batch,
                                                float* __restrict__ ws) {
    __shared__ float tile[WPB * TILE_R * LDS_PITCH];
    __shared__ float s_sum[BG], s_sq[BG], s_cnt[BSEG], s_touch[BSEG];

    float* gsum = ws;
    float* gsq  = ws + BG;
    float* gcnt = ws + 2 * BG;

    for (int t = threadIdx.x; t < BG; t += 256) { s_sum[t] = 0.0f; s_sq[t] = 0.0f; }
    if (threadIdx.x < BSEG) { s_cnt[threadIdx.x] = 0.0f; s_touch[threadIdx.x] = 0.0f; }
    __syncthreads();

    const int lane = threadIdx.x & 31;
    const int wave = threadIdx.x >> 5;
    const int m    = lane & 15;   // row (A layout) / N column (D layout)
    const int half = lane >> 4;   // 0: lanes 0-15, 1: lanes 16-31
    float* slice = tile + wave * (TILE_R * LDS_PITCH);
    const unsigned ldsBase = (unsigned)(uintptr_t)slice;  // low 32b = LDS offset

    const int nTiles = N_ROWS / TILE_R;  // N divisible by 16
    const int T = nTiles * G;            // one task = (16-row tile, group)
    const int gw0 = blockIdx.x * WPB + wave;
    const int totalWaves = gridDim.x * WPB;
    const int chunk = (T + totalWaves - 1) / totalWaves;
    const int tBegin = gw0 * chunk;
    const int tEnd   = (tBegin + chunk < T) ? (tBegin + chunk) : T;

    for (int task = tBegin; task < tEnd; ++task) {
        const int g  = task & (G - 1);
        const int r0 = (task >> 2) * TILE_R;

        // ---- async stage: 16 rows x 64 f32 (256B/row) into this wave's slice
        #pragma unroll
        for (int j = 0; j < 8; ++j) {
            const int rr = 2 * j + half;  // lanes 0-15: row 2j, 16-31: row 2j+1
            const float* gp = x + (size_t)(r0 + rr) * CCH + g * GC + m * 4;
            const unsigned la = ldsBase + (unsigned)(rr * LDS_PITCH + m * 4) * 4u;
            asm volatile("global_load_async_to_lds_b128 %0, %1, off"
                         :: "v"(la), "v"(gp) : "memory");
        }
        asm volatile("s_wait_asynccnt 0" ::: "memory");

#ifdef HAVE_WMMA_F32X4
        // ---- WMMA row-reduction: D = A(16x4) * ones(4x16) accumulated over K
        v8f acc  = {};
        v8f acc2 = {};
        const v2f onesv = {1.0f, 1.0f};
        #pragma unroll
        for (int k = 0; k < 16; ++k) {
            // A layout: lane m holds row m; VGPR0/1 = K {2*half, 2*half+1}
            const float2 av = *reinterpret_cast<const float2*>(
                slice + m * LDS_PITCH + (k << 2) + 2 * half);
            v2f a; a.x = av.x; a.y = av.y;
            acc = __builtin_amdgcn_wmma_f32_16x16x4_f32(
                false, a, false, onesv, (short)0, acc, false, false);
            const v2f a2 = a * a;
            acc2 = __builtin_amdgcn_wmma_f32_16x16x4_f32(
                false, a2, false, onesv, (short)0, acc2, false, false);
        }
        // D layout: VGPR r -> row r (lanes 0-15), row 8+r (lanes 16-31)
        if (m == 0) {
            const int rbase = r0 + half * 8;
            #pragma unroll
            for (int r = 0; r < 8; ++r) {
                const int row = rbase + r;
                const int b = batch[row];
                s_touch[b] = 1.0f;
                atomicAdd(&s_sum[b * G + g], acc[r]);
                atomicAdd(&s_sq [b * G + g], acc2[r]);
                if (g == 0) atomicAdd(&s_cnt[b], 1.0f);
            }
        }
#else
        // ---- fallback: lane m owns row m, halves split the 64 cols, shfl-combine
        float s = 0.0f, q = 0.0f;
        #pragma unroll
        for (int i = 0; i < 32; ++i) {
            const float v = slice[m * LDS_PITCH + half * 32 + i];
            s += v; q += v * v;
        }
        s += __shfl_down(s, 16);
        q += __shfl_down(q, 16);
        if (half == 0) {
            const int row = r0 + m;
            const int b = batch[row];
            s_touch[b] = 1.0f;
            atomicAdd(&s_sum[b * G + g], s);
            atomicAdd(&s_sq [b * G + g], q);
            if (g == 0) atomicAdd(&s_cnt[b], 1.0f);
        }
#endif
    }

    __syncthreads();
    for (int t = threadIdx.x; t < BG; t += 256) {
        if (s_touch[t / G] != 0.0f) {
            atomicAdd(&gsum[t], s_sum[t]);
            atomicAdd(&gsq[t],  s_sq[t]);
        }
    }
    if (threadIdx.x < BSEG && s_cnt[threadIdx.x] != 0.0f)
        atomicAdd(&gcnt[threadIdx.x], s_cnt[threadIdx.x]);
}

__global__ void gn_finalize(float* __restrict__ ws) {
    const int t = threadIdx.x;
    if (t < BG) {
        const float deg  = ws[2 * BG + t / G];
        const float norm = fmaxf(deg, 1.0f) * (float)GC;
        const float mean = ws[t] / norm;
        float var = ws[BG + t] / norm - mean * mean;
        var = fmaxf(var, 0.0f);
        ws[2 * BG + BSEG + t]      = mean;
        ws[2 * BG + BSEG + BG + t] = rsqrtf(var + EPS_F);
    }
}

__global__ __launch_bounds__(256) void gn_apply(const float* __restrict__ x,
                                                const int* __restrict__ batch,
                                                const float* __restrict__ weight,
                                                const float* __restrict__ bias,
                                                const float* __restrict__ ws,
                                                float* __restrict__ out) {
    __shared__ float s_w[CCH], s_b[CCH], s_mean[BG], s_rstd[BG];
    const int t = threadIdx.x;
    s_w[t]    = weight[t];
    s_b[t]    = bias[t];
    s_mean[t] = ws[2 * BG + BSEG + t];
    s_rstd[t] = ws[2 * BG + BSEG + BG + t];
    __syncthreads();

    const int total = N_ROWS * (CCH / 4);  // 32M float4, fits int
    const float4* xv = (const float4*)x;
    float4*       ov = (float4*)out;
    const int stride = gridDim.x * 256;

    for (int i = blockIdx.x * 256 + t; i < total; i += stride) {
        const int row = i >> 6;        // 64 float4 per row
        const int q   = i & 63;
        const int b   = batch[row];
        const int idx = b * G + (q >> 4);
        const float mean = s_mean[idx];
        const float rstd = s_rstd[idx];
        const float4 v  = xv[i];
        const float4 w  = ((const float4*)s_w)[q];
        const float4 bb = ((const float4*)s_b)[q];
        float4 o;
        o.x = (v.x - mean) * rstd * w.x + bb.x;
        o.y = (v.y - mean) * rstd * w.y + bb.y;
        o.z = (v.z - mean) * rstd * w.z + bb.z;
        o.w = (v.w - mean) * rstd * w.w + bb.w;
        ov[i] = o;
    }
}

extern "C" void kernel_launch(void* const* d_in, const int* in_sizes, int n_in,
                              void* d_out, int out_size, void* d_ws, size_t ws_size,
                              hipStream_t stream) {
    const float* x      = (const float*)d_in[0];
    const int*   batch  = (const int*)d_in[1];
    const float* weight = (const float*)d_in[2];
    const float* bias   = (const float*)d_in[3];
    float* out = (float*)d_out;
    float* ws  = (float*)d_ws;

    gn_zero<<<1, 1024, 0, stream>>>(ws);
    gn_stats<<<1024, 256, 0, stream>>>(x, batch, ws);
    gn_finalize<<<1, 256, 0, stream>>>(ws);
    gn_apply<<<4096, 256, 0, stream>>>(x, batch, weight, bias, ws, out);
}